// HalfCDLoss_44813688767320
// MI455X (gfx1250) — compile-verified
//
#include <hip/hip_runtime.h>

typedef __attribute__((ext_vector_type(2))) float v2f;
typedef __attribute__((ext_vector_type(8))) float v8f;

#define BATCH 4
#define NPTS 8192
#define WAVES_PER_BATCH (NPTS / 16)     // 512 waves cover 8192 prediction rows
#define CHUNK_TILES 64                  // gt tiles staged in LDS per pass
#define CHUNK_PTS (CHUNK_TILES * 16)    // 1024 gt points per chunk
#define NCHUNKS (NPTS / CHUNK_PTS)      // 8 chunks

// One wave owns 16 prediction points. The block (8 waves, all same batch)
// cooperatively stages the WMMA-ready B operand for 64 gt tiles into LDS:
//   B col (per gt point) = [-2gx, -2gy, -2gz, ||g||^2]  (lane-ready v2f pairs)
// Steady-state inner loop per 2 tiles:
//   1x ds_load_2addr_b64 (prefetch next pair) -> 2x v_wmma_f32_16x16x4_f32
//   -> 8x v_min3_num_f32 (folds both D tiles into the row-min accumulators).
// A row = [px,py,pz,1] => D[r][n] = -2 p.g + ||g||^2; +||p||^2 and clamp are
// applied after the min (both commute with it).
__global__ __launch_bounds__(256) void cd_wmma_kernel(const float* __restrict__ pred,
                                                      const float* __restrict__ gt,
                                                      float* __restrict__ ws) {
    // +2 pad tiles: the last prefetch reads (unused) garbage instead of OOB.
    __shared__ v2f ldsB[(CHUNK_TILES + 2) * 32];   // 16.5 KB

    const int tid   = threadIdx.x;
    const int lane  = tid & 31;
    const int gwave = blockIdx.x * 8 + (tid >> 5);
    const int b     = gwave / WAVES_PER_BATCH;
    const int wib   = gwave % WAVES_PER_BATCH;
    const int col   = lane & 15;
    const bool upper = lane >= 16;

    // Load this lane's prediction point (row = col of the A tile).
    const size_t pbase = ((size_t)b * NPTS + (size_t)wib * 16 + col) * 3;
    const float px = pred[pbase + 0];
    const float py = pred[pbase + 1];
    const float pz = pred[pbase + 2];

    // A-matrix VGPR layout: VGPR0 = K0/K2 (lower/upper half), VGPR1 = K1/K3.
    v2f a;
    a.x = upper ? pz   : px;   // K2 : K0
    a.y = upper ? 1.0f : py;   // K3 : K1
    const float p2 = px * px + py * py + pz * pz;

    float minv[8];
#pragma unroll
    for (int i = 0; i < 8; ++i) minv[i] = 3.402823466e38f;

    const float* gbase = gt + (size_t)b * NPTS * 3;

    for (int c = 0; c < NCHUNKS; ++c) {
        // ---- Stage: 256 threads precompute 1024 gt points' B operands ----
#pragma unroll
        for (int j = 0; j < CHUNK_PTS / 256; ++j) {
            const int p  = j * 256 + tid;          // 0..1023 within chunk
            const float* g = gbase + (size_t)(c * CHUNK_PTS + p) * 3;
            const float gx = g[0], gy = g[1], gz = g[2];
            const float g2 = gx * gx + gy * gy + gz * gz;
            const int t  = p >> 4;
            const int cc = p & 15;
            v2f lo, hi;
            lo.x = -2.0f * gx; lo.y = -2.0f * gy;   // lanes 0-15: K0,K1
            hi.x = -2.0f * gz; hi.y = g2;           // lanes 16-31: K2,K3
            ldsB[t * 32 + cc]      = lo;
            ldsB[t * 32 + 16 + cc] = hi;
        }
        __syncthreads();

        // ---- Compute: 64 WMMA tiles, 2 per iteration, 1-stage pipelined ----
        v2f bm0 = ldsB[lane];
        v2f bm1 = ldsB[32 + lane];
#pragma unroll 4
        for (int t = 0; t < CHUNK_TILES; t += 2) {
            // Prefetch next tile pair (merges into one ds_load_2addr_b64).
            v2f bn0 = ldsB[(t + 2) * 32 + lane];
            v2f bn1 = ldsB[(t + 3) * 32 + lane];
            v8f cz = {0.0f, 0.0f, 0.0f, 0.0f, 0.0f, 0.0f, 0.0f, 0.0f};
            v8f d0 = __builtin_amdgcn_wmma_f32_16x16x4_f32(
                false, a, false, bm0, (short)0, cz, false, false);
            v8f d1 = __builtin_amdgcn_wmma_f32_16x16x4_f32(
                false, a, false, bm1, (short)0, cz, false, false);
            // 3-source min folds both D tiles per accumulator; raw asm avoids
            // clang's sNaN-quieting max(x,x) canonicalization.
#pragma unroll
            for (int i = 0; i < 8; ++i)
                asm("v_min3_num_f32 %0, %0, %1, %2"
                    : "+v"(minv[i]) : "v"(d0[i]), "v"(d1[i]));
            bm0 = bn0;
            bm1 = bn1;
        }
        __syncthreads();
    }

    // Column-min across the 16 lanes of each half-wave (masks 1..8 stay in-half).
#pragma unroll
    for (int m = 1; m <= 8; m <<= 1) {
#pragma unroll
        for (int i = 0; i < 8; ++i) {
            float o = __shfl_xor(minv[i], m, 32);
            asm("v_min_num_f32 %0, %0, %1" : "+v"(minv[i]) : "v"(o));
        }
    }

    // Route each lane the min of its own row r = col.
    // Lower-half lanes hold rows 0..7 in minv[0..7]; upper-half lanes rows 8..15.
    const int idx = col & 7;
    float sel = minv[0];
#pragma unroll
    for (int i = 1; i < 8; ++i) sel = (idx == i) ? minv[i] : sel;
    const float other   = __shfl_xor(sel, 16, 32);
    const bool rowUpper = (col >> 3) != 0;
    const float rowmin  = (rowUpper == upper) ? sel : other;

    float dmin = fmaxf(rowmin + p2, 0.0f);
    // Sum the 16 rows (upper half duplicates lower half).
#pragma unroll
    for (int m = 1; m <= 8; m <<= 1) dmin += __shfl_xor(dmin, m, 32);

    if (lane == 0) ws[(size_t)b * WAVES_PER_BATCH + wib] = dmin;
}

// Deterministic tree reduction of the 512 per-wave partials per batch.
__global__ __launch_bounds__(256) void cd_reduce_kernel(const float* __restrict__ ws,
                                                        float* __restrict__ out) {
    __shared__ float s[256];
    const int b   = blockIdx.x;
    const int tid = threadIdx.x;
    s[tid] = ws[b * WAVES_PER_BATCH + tid] + ws[b * WAVES_PER_BATCH + tid + 256];
    __syncthreads();
#pragma unroll
    for (int off = 128; off > 0; off >>= 1) {
        if (tid < off) s[tid] += s[tid + off];
        __syncthreads();
    }
    if (tid == 0) out[b] = s[0] * (1.0f / (float)NPTS);
}

extern "C" void kernel_launch(void* const* d_in, const int* in_sizes, int n_in,
                              void* d_out, int out_size, void* d_ws, size_t ws_size,
                              hipStream_t stream) {
    const float* pred = (const float*)d_in[0];
    const float* gt   = (const float*)d_in[1];
    float* out = (float*)d_out;
    float* ws  = (float*)d_ws;   // needs 4*512*4 = 8 KB

    // 2048 waves total: 256 blocks x 8 waves; 512 waves per batch;
    // all 8 waves of a block share one batch (512 % 8 == 0).
    cd_wmma_kernel<<<256, 256, 0, stream>>>(pred, gt, ws);
    cd_reduce_kernel<<<BATCH, 256, 0, stream>>>(ws, out);
}